// GCN_5471788335177
// MI455X (gfx1250) — compile-verified
//
#include <hip/hip_runtime.h>

typedef float v2f __attribute__((ext_vector_type(2)));
typedef float v8f __attribute__((ext_vector_type(8)));

#define HIDF 128

__global__ void init_deg_kernel(float* __restrict__ deg, int n) {
    int i = blockIdx.x * blockDim.x + threadIdx.x;
    if (i < n) deg[i] = 1.0f;   // self-loop contributes 1 to every node's degree
}

__global__ void count_deg_kernel(const int* __restrict__ dst, float* __restrict__ deg, int e) {
    int i = blockIdx.x * blockDim.x + threadIdx.x;
    if (i < e) atomicAdd(&deg[dst[i]], 1.0f);
}

__global__ void dinv_kernel(float* __restrict__ deg, int n) {
    int i = blockIdx.x * blockDim.x + threadIdx.x;
    if (i < n) deg[i] = rsqrtf(deg[i]);   // deg >= 1 always (self-loops)
}

__global__ void zero_kernel(float* __restrict__ p, long n) {
    long i = (long)blockIdx.x * blockDim.x + threadIdx.x;
    if (i < n) p[i] = 0.0f;
}

// Out[M x 128] = H[M x 128] @ W[128 x 128] via V_WMMA_F32_16X16X4_F32.
// One wave computes a 16x128 slab: 8 f32 accumulators (v8f), K-loop of 32 steps,
// A-fragment reused across the 8 column tiles.
__global__ void gemm_wmma_kernel(const float* __restrict__ H,
                                 const float* __restrict__ W,
                                 float* __restrict__ Out, int M) {
    int wave  = threadIdx.x >> 5;
    int lane  = threadIdx.x & 31;
    int mtile = blockIdx.x * 4 + wave;
    if (mtile * 16 >= M) return;          // wave-uniform: EXEC stays all-ones
    int half = lane >> 4;                 // 0 or 1
    int lm   = lane & 15;                 // 0..15

    const float* hrow = H + (long)(mtile * 16 + lm) * HIDF;

    v8f c[8];
#pragma unroll
    for (int i = 0; i < 8; ++i) c[i] = (v8f){0.f, 0.f, 0.f, 0.f, 0.f, 0.f, 0.f, 0.f};

    for (int k0 = 0; k0 < HIDF; k0 += 4) {
        // A fragment (16x4, f32): lane holds row lm, K = k0 + 2*half + {0,1}
        v2f a = *(const v2f*)(hrow + k0 + 2 * half);
        const float* wk = W + (long)(k0 + 2 * half) * HIDF + lm;
#pragma unroll
        for (int nt = 0; nt < 8; ++nt) {
            // B fragment (4x16, f32): lane holds col lm, K = k0 + 2*half + {0,1}
            v2f b;
            b.x = wk[nt * 16];
            b.y = wk[HIDF + nt * 16];
            c[nt] = __builtin_amdgcn_wmma_f32_16x16x4_f32(
                false, a, false, b, (short)0, c[nt], false, false);
        }
    }

#pragma unroll
    for (int nt = 0; nt < 8; ++nt) {
#pragma unroll
        for (int i = 0; i < 8; ++i) {
            // C/D layout: VGPR i -> row i + 8*half, col = lm within tile nt
            Out[(long)(mtile * 16 + i + 8 * half) * HIDF + nt * 16 + lm] = c[nt][i];
        }
    }
}

// One wave per edge: float4-coalesced gather of hw[src], scaled atomic scatter to agg[dst].
__global__ void scatter_kernel(const float* __restrict__ hw,
                               const int* __restrict__ src,
                               const int* __restrict__ dst,
                               const float* __restrict__ dinv,
                               float* __restrict__ agg, int e) {
    int edge = blockIdx.x * (blockDim.x >> 5) + (threadIdx.x >> 5);
    int lane = threadIdx.x & 31;
    if (edge >= e) return;
    int s = src[edge];
    int d = dst[edge];
    float norm = dinv[s] * dinv[d];
    float4 v = *((const float4*)(hw + (long)s * HIDF) + lane);
    float* out = agg + (long)d * HIDF + lane * 4;
    atomicAdd(out + 0, v.x * norm);
    atomicAdd(out + 1, v.y * norm);
    atomicAdd(out + 2, v.z * norm);
    atomicAdd(out + 3, v.w * norm);
}

// Fused epilogue: h = relu(agg + hw*dinv^2 (self-loop) + bias). Stream order makes
// the self-loop add race-free without atomics.
__global__ void finish_kernel(const float* __restrict__ hw,
                              const float* __restrict__ dinv,
                              const float* __restrict__ bias,
                              float* __restrict__ h, long total) {
    long i = (long)blockIdx.x * blockDim.x + threadIdx.x;
    if (i >= total) return;
    int node = (int)(i >> 7);
    int f    = (int)(i & 127);
    float di = dinv[node];
    float v  = h[i] + hw[i] * di * di + bias[f];
    h[i] = v > 0.0f ? v : 0.0f;
}

// One block (128 threads) per graph; thread t owns feature t; mean over node range.
__global__ void pool_kernel(const float* __restrict__ h,
                            const int* __restrict__ ptr,
                            float* __restrict__ out, int g) {
    int gi = blockIdx.x;
    int t  = threadIdx.x;
    int s  = ptr[gi];
    int e  = ptr[gi + 1];
    float acc = 0.0f;
    for (int n = s; n < e; ++n) acc += h[(long)n * HIDF + t];
    float cnt = (float)(e - s);
    if (cnt < 1.0f) cnt = 1.0f;
    out[(long)gi * HIDF + t] = acc / cnt;
}

extern "C" void kernel_launch(void* const* d_in, const int* in_sizes, int n_in,
                              void* d_out, int out_size, void* d_ws, size_t ws_size,
                              hipStream_t stream) {
    const float* x   = (const float*)d_in[0];
    const int*   ei  = (const int*)d_in[1];
    const int*   ptr = (const int*)d_in[2];
    const float* Wl[3] = {(const float*)d_in[3], (const float*)d_in[5], (const float*)d_in[7]};
    const float* bl[3] = {(const float*)d_in[4], (const float*)d_in[6], (const float*)d_in[8]};

    int N = in_sizes[0] / HIDF;
    int E = in_sizes[1] / 2;
    int G = in_sizes[2] - 1;

    const int* src = ei;       // edge_index[0]
    const int* dst = ei + E;   // edge_index[1]

    // Workspace carve-out: dinv[N] | bufA[N*128] | bufB[N*128]
    float* dinv = (float*)d_ws;
    size_t npad = ((size_t)N + 255) & ~(size_t)255;
    float* bufA = dinv + npad;
    float* bufB = bufA + (size_t)N * HIDF;
    long total  = (long)N * HIDF;

    init_deg_kernel<<<(N + 255) / 256, 256, 0, stream>>>(dinv, N);
    count_deg_kernel<<<(E + 255) / 256, 256, 0, stream>>>(dst, dinv, E);
    dinv_kernel<<<(N + 255) / 256, 256, 0, stream>>>(dinv, N);

    int mtiles      = (N + 15) / 16;
    int gemm_blocks = (mtiles + 3) / 4;
    int scat_blocks = (E + 7) / 8;
    int elem_blocks = (int)((total + 255) / 256);

    const float* hin = x;
    for (int l = 0; l < 3; ++l) {
        gemm_wmma_kernel<<<gemm_blocks, 128, 0, stream>>>(hin, Wl[l], bufA, N);
        zero_kernel<<<elem_blocks, 256, 0, stream>>>(bufB, total);
        scatter_kernel<<<scat_blocks, 256, 0, stream>>>(bufA, src, dst, dinv, bufB, E);
        finish_kernel<<<elem_blocks, 256, 0, stream>>>(bufA, dinv, bl[l], bufB, total);
        hin = bufB;
    }

    pool_kernel<<<G, HIDF, 0, stream>>>(bufB, ptr, (float*)d_out, G);
}